// GIPA2Para_34119220199762
// MI455X (gfx1250) — compile-verified
//
#include <hip/hip_runtime.h>
#include <hip/hip_bf16.h>

typedef float v2f __attribute__((ext_vector_type(2)));
typedef float v8f __attribute__((ext_vector_type(8)));
typedef int v4i_b __attribute__((vector_size(16)));   // matches builtin's param pointee

#define HDIM 256
#define NEG_SLOPE_C 0.2f
#define CLAMP_MIN_C 1e-9f
#define GN_EPS_C 1e-9f
#define BN_EPS_C 1e-5f

#define ASG __attribute__((address_space(1)))
#define ASL __attribute__((address_space(3)))

#if defined(__has_builtin)
#if __has_builtin(__builtin_amdgcn_global_load_async_to_lds_b128)
#define HAVE_ASYNC_B128 1
#endif
#endif
#ifndef HAVE_ASYNC_B128
#define HAVE_ASYNC_B128 0
#endif

static __device__ __forceinline__ void wait_async0() {
#if defined(__has_builtin)
#if __has_builtin(__builtin_amdgcn_s_wait_asynccnt)
    __builtin_amdgcn_s_wait_asynccnt(0);
    return;
#endif
#endif
    asm volatile("s_wait_asynccnt 0x0" ::: "memory");
}

// ---------------- order-preserving float<->uint for atomicMax on floats -------------
static __device__ __forceinline__ unsigned f2ord(float f) {
    unsigned u = __float_as_uint(f);
    return (u & 0x80000000u) ? ~u : (u | 0x80000000u);
}
static __device__ __forceinline__ float ord2f(unsigned u) {
    return (u & 0x80000000u) ? __uint_as_float(u & 0x7FFFFFFFu)
                             : __uint_as_float(~u);
}

// ---------------- K=8 GEMM + bias + relu (encoder [N,8]x[8,256], edge emb [E,8]x[8,16])
__global__ __launch_bounds__(256) void gemm_k8_relu_kernel(
    const float* __restrict__ X, const float* __restrict__ W,
    const float* __restrict__ bias, float* __restrict__ Y, int M, int NC)
{
    const int lane = threadIdx.x & 31;
    const int wave = threadIdx.x >> 5;
    const int ntiles = NC >> 4;
    const int tile = blockIdx.x * 8 + wave;
    const int total = (M >> 4) * ntiles;
    if (tile >= total) return;                 // whole-wave exit: EXEC stays all-1s
    const int m0 = (tile / ntiles) << 4;
    const int n0 = (tile % ntiles) << 4;
    const int mm = lane & 15;
    const int nn = lane & 15;
    const int kh2 = (lane >> 4) << 1;          // 0 or 2 (K-half per A/B layout)

    v8f c = {};
#pragma unroll
    for (int k0 = 0; k0 < 8; k0 += 4) {
        v2f a, b;
        a.x = X[(size_t)(m0 + mm) * 8 + k0 + kh2];
        a.y = X[(size_t)(m0 + mm) * 8 + k0 + kh2 + 1];
        b.x = W[(size_t)(k0 + kh2) * NC + n0 + nn];
        b.y = W[(size_t)(k0 + kh2 + 1) * NC + n0 + nn];
        c = __builtin_amdgcn_wmma_f32_16x16x4_f32(false, a, false, b,
                                                  (short)0, c, false, false);
    }
    const float bv = bias[n0 + nn];
#pragma unroll
    for (int r = 0; r < 8; ++r) {
        const int row = m0 + r + ((lane >> 4) << 3);
        Y[(size_t)row * NC + n0 + nn] = fmaxf(c[r] + bv, 0.0f);
    }
}

// ---------------- K=256 GEMM: Y = X[M,256] @ W[256,256] (+bias)(+add1)(+add2) --------
// Block = 256 threads (8 waves). Each block computes a 32x128 output tile:
// each wave owns one 16-wide N-tile and TWO M-tiles, so every B fragment
// (loaded from L2) feeds two v_wmma ops. A tile (32x256) staged in LDS via
// async-to-LDS B128 copies when available.
__global__ __launch_bounds__(256) void gemm_k256_kernel(
    const float* __restrict__ X, const float* __restrict__ W,
    const float* __restrict__ bias, const float* __restrict__ add1,
    const float* __restrict__ add2, float* __restrict__ Y, int M)
{
    __shared__ float As[32][260];              // pad 256->260 => conflict-free frag reads
    const int lane = threadIdx.x & 31;
    const int wave = threadIdx.x >> 5;
    const int m0 = blockIdx.y << 5;
    const int n0 = ((blockIdx.x << 3) + wave) << 4;

    {   // stage 32x256 A tile cooperatively
        const int row = threadIdx.x >> 3;          // 0..31
        const int cb  = (threadIdx.x & 7) << 5;    // 0,32,...,224
        const float* g = X + (size_t)(m0 + row) * HDIM + cb;
#if HAVE_ASYNC_B128
#pragma unroll
        for (int j = 0; j < 8; ++j)
            __builtin_amdgcn_global_load_async_to_lds_b128(
                (ASG v4i_b*)(g + 4 * j),
                (ASL v4i_b*)&As[row][cb + 4 * j], 0, 0);
        wait_async0();
#else
        const float4* s4 = (const float4*)g;
        float4* d4 = (float4*)&As[row][cb];
#pragma unroll
        for (int j = 0; j < 8; ++j) d4[j] = s4[j];
#endif
    }
    __syncthreads();

    const int mm = lane & 15;
    const int nn = lane & 15;
    const int kh2 = (lane >> 4) << 1;
    const float* Wp = W + n0 + nn;

    v8f c0 = {}, c1 = {};
#pragma unroll 8
    for (int k0 = 0; k0 < HDIM; k0 += 4) {
        if ((k0 & 31) == 0 && k0 + 128 < HDIM)     // stream W slab ahead into caches
            __builtin_prefetch(Wp + (size_t)(k0 + 128) * HDIM, 0, 1);
        v2f a0, a1, b;
        b.x  = Wp[(size_t)(k0 + kh2) * HDIM];
        b.y  = Wp[(size_t)(k0 + kh2 + 1) * HDIM];
        a0.x = As[mm][k0 + kh2];
        a0.y = As[mm][k0 + kh2 + 1];
        a1.x = As[mm + 16][k0 + kh2];
        a1.y = As[mm + 16][k0 + kh2 + 1];
        c0 = __builtin_amdgcn_wmma_f32_16x16x4_f32(false, a0, false, b,
                                                   (short)0, c0, false, false);
        c1 = __builtin_amdgcn_wmma_f32_16x16x4_f32(false, a1, false, b,
                                                   (short)0, c1, false, false);
    }

    const int nidx = n0 + nn;
    const float bv = bias ? bias[nidx] : 0.0f;
#pragma unroll
    for (int r = 0; r < 8; ++r) {
        const int row = m0 + r + ((lane >> 4) << 3);
        {
            const size_t idx = (size_t)row * HDIM + nidx;
            float v = c0[r] + bv;
            if (add1) v += add1[idx];
            if (add2) v += add2[idx];
            Y[idx] = v;
        }
        {
            const size_t idx = (size_t)(row + 16) * HDIM + nidx;
            float v = c1[r] + bv;
            if (add1) v += add1[idx];
            if (add2) v += add2[idx];
            Y[idx] = v;
        }
    }
}

// ---------------- edge logit recompute (leaky_relu(attn_s[s]+attn_d[d]+ee@aew)) ------
static __device__ __forceinline__ float edge_logit(
    int e, int ch, const int* __restrict__ src, const int* __restrict__ dst,
    const float* __restrict__ attn_s, const float* __restrict__ attn_d,
    const float* __restrict__ ee, const float* __restrict__ aew,
    int* s_out, int* d_out)
{
    const int s = src[e], d = dst[e];
    float acc = attn_s[(size_t)s * HDIM + ch] + attn_d[(size_t)d * HDIM + ch];
    const float* er = ee + (size_t)e * 16;
#pragma unroll
    for (int k = 0; k < 16; ++k)
        acc = fmaf(er[k], aew[(size_t)k * HDIM + ch], acc);
    *s_out = s; *d_out = d;
    return acc >= 0.0f ? acc : NEG_SLOPE_C * acc;
}

__global__ __launch_bounds__(256) void seg_max_kernel(
    const int* __restrict__ src, const int* __restrict__ dst,
    const float* __restrict__ attn_s, const float* __restrict__ attn_d,
    const float* __restrict__ ee, const float* __restrict__ aew,
    unsigned* __restrict__ mdst, unsigned* __restrict__ msrc)
{
    const int e = blockIdx.x, ch = threadIdx.x;
    int s, d;
    const float v = edge_logit(e, ch, src, dst, attn_s, attn_d, ee, aew, &s, &d);
    const unsigned o = f2ord(v);
    atomicMax(&mdst[(size_t)d * HDIM + ch], o);
    atomicMax(&msrc[(size_t)s * HDIM + ch], o);
}

__global__ __launch_bounds__(256) void seg_sum_kernel(
    const int* __restrict__ src, const int* __restrict__ dst,
    const float* __restrict__ attn_s, const float* __restrict__ attn_d,
    const float* __restrict__ ee, const float* __restrict__ aew,
    const unsigned* __restrict__ mdst, const unsigned* __restrict__ msrc,
    float* __restrict__ sdst, float* __restrict__ ssrc)
{
    const int e = blockIdx.x, ch = threadIdx.x;
    int s, d;
    const float v = edge_logit(e, ch, src, dst, attn_s, attn_d, ee, aew, &s, &d);
    const float md = ord2f(mdst[(size_t)d * HDIM + ch]);
    const float ms = ord2f(msrc[(size_t)s * HDIM + ch]);
    atomicAdd(&sdst[(size_t)d * HDIM + ch], expf(v - md));
    atomicAdd(&ssrc[(size_t)s * HDIM + ch], expf(v - ms));
}

__global__ __launch_bounds__(256) void message_kernel(
    const int* __restrict__ src, const int* __restrict__ dst,
    const float* __restrict__ attn_s, const float* __restrict__ attn_d,
    const float* __restrict__ ee, const float* __restrict__ aew,
    const unsigned* __restrict__ mdst, const unsigned* __restrict__ msrc,
    const float* __restrict__ sdst, const float* __restrict__ ssrc,
    const float* __restrict__ fsrc, float* __restrict__ msg)
{
    const int e = blockIdx.x, ch = threadIdx.x;
    int s, d;
    const float v = edge_logit(e, ch, src, dst, attn_s, attn_d, ee, aew, &s, &d);
    const size_t di = (size_t)d * HDIM + ch, si = (size_t)s * HDIM + ch;
    const float pd = expf(v - ord2f(mdst[di])) / sdst[di];
    const float ps = expf(v - ord2f(msrc[si])) / ssrc[si];
    const float a = sqrtf(fmaxf(pd, CLAMP_MIN_C) * fmaxf(ps, CLAMP_MIN_C));
    atomicAdd(&msg[di], a * fsrc[si]);
}

// ---------------- per-head GroupNorm: one node per block, one head per wave32 --------
__global__ __launch_bounds__(256) void groupnorm_kernel(
    float* __restrict__ msg, const float* __restrict__ gscale,
    const float* __restrict__ goff)
{
    const int n = blockIdx.x, ch = threadIdx.x;
    const size_t idx = (size_t)n * HDIM + ch;
    const float v = msg[idx];
    float s = v, s2 = v * v;
#pragma unroll
    for (int off = 16; off > 0; off >>= 1) {
        s  += __shfl_xor(s, off, 32);
        s2 += __shfl_xor(s2, off, 32);
    }
    const float mean = s * (1.0f / 32.0f);
    const float var = s2 * (1.0f / 32.0f) - mean * mean + GN_EPS_C;
    msg[idx] = (v - mean) * gscale[ch] * rsqrtf(var) + goff[ch];
}

// ---------------- BatchNorm over nodes ----------------------------------------------
__global__ __launch_bounds__(256) void bn_stats_kernel(
    const float* __restrict__ x, float* __restrict__ bsum,
    float* __restrict__ bsq, int N)
{
    const int ch = threadIdx.x;
    float s = 0.0f, s2 = 0.0f;
    for (int n = blockIdx.x; n < N; n += gridDim.x) {
        const float v = x[(size_t)n * HDIM + ch];
        s += v; s2 += v * v;
    }
    atomicAdd(&bsum[ch], s);
    atomicAdd(&bsq[ch], s2);
}

__global__ __launch_bounds__(256) void bn_apply_kernel(
    const float* __restrict__ x, const float* __restrict__ bsum,
    const float* __restrict__ bsq, const float* __restrict__ gamma,
    const float* __restrict__ beta, float* __restrict__ out, int N)
{
    const int ch = threadIdx.x;
    const float inv = 1.0f / (float)N;
    const float mu = bsum[ch] * inv;
    const float var = bsq[ch] * inv - mu * mu;
    const float r = rsqrtf(var + BN_EPS_C) * gamma[ch];
    const float b = beta[ch];
    for (int n = blockIdx.x; n < N; n += gridDim.x) {
        const size_t idx = (size_t)n * HDIM + ch;
        out[idx] = fmaxf((x[idx] - mu) * r + b, 0.0f);
    }
}

__global__ void fill_u32_kernel(unsigned* __restrict__ p, unsigned v, size_t n) {
    size_t i = (size_t)blockIdx.x * blockDim.x + threadIdx.x;
    const size_t stride = (size_t)gridDim.x * blockDim.x;
    for (; i < n; i += stride) p[i] = v;
}

// ====================================================================================
extern "C" void kernel_launch(void* const* d_in, const int* in_sizes, int n_in,
                              void* d_out, int out_size, void* d_ws, size_t ws_size,
                              hipStream_t stream)
{
    (void)n_in; (void)out_size; (void)ws_size;
    const float* feat       = (const float*)d_in[0];
    const float* efeat      = (const float*)d_in[1];
    const float* node_w     = (const float*)d_in[2];
    const float* node_b     = (const float*)d_in[3];
    const float* edge_w     = (const float*)d_in[4];
    const float* edge_b     = (const float*)d_in[5];
    const float* src_w      = (const float*)d_in[6];
    const float* dst_w      = (const float*)d_in[7];
    const float* dst_b      = (const float*)d_in[8];
    const float* attn_src_w = (const float*)d_in[9];
    const float* attn_dst_w = (const float*)d_in[10];
    const float* attn_edge_w= (const float*)d_in[11];
    const float* gn_scale   = (const float*)d_in[12];
    const float* gn_offset  = (const float*)d_in[13];
    const float* agg_w      = (const float*)d_in[14];
    const float* agg_b      = (const float*)d_in[15];
    const float* bn_gamma   = (const float*)d_in[16];
    const float* bn_beta    = (const float*)d_in[17];
    const int*   src        = (const int*)d_in[18];
    const int*   dst        = (const int*)d_in[19];

    const int N = in_sizes[0] / 8;
    const int E = in_sizes[1] / 8;
    const int L = 2;
    const size_t NH = (size_t)N * HDIM;

    // workspace carve-up (all node tensors ~20 MB -> L2 resident on MI455X)
    float* ws     = (float*)d_ws;
    float* h      = ws;                 // [N,256] post-BN activations
    float* hlast  = h      + NH;        // [N,256] pre-BN residual
    float* attn_s = hlast  + NH;
    float* attn_d = attn_s + NH;
    float* fsrc   = attn_d + NH;
    float* dterm  = fsrc   + NH;
    float* msg    = dterm  + NH;        // msg .. ssrc are contiguous (zeroed together)
    unsigned* mdst = (unsigned*)(msg + NH);
    unsigned* msrc = mdst + NH;
    float* sdst   = (float*)(msrc + NH);
    float* ssrc   = sdst + NH;
    float* ee     = ssrc + NH;          // [E,16]
    float* bsum   = ee + (size_t)E * 16;
    float* bsq    = bsum + HDIM;

    const dim3 blk(256);
    const dim3 gemm256_grid(2, N >> 5);   // 32 output rows per block

    // node encoder: h = relu(feat @ node_w + node_b)
    {
        const int total = (N >> 4) * (HDIM >> 4);
        gemm_k8_relu_kernel<<<(total + 7) / 8, blk, 0, stream>>>(
            feat, node_w, node_b, h, N, HDIM);
    }

    for (int i = 0; i < L; ++i) {
        const float* ew  = edge_w      + (size_t)i * 8 * 16;
        const float* eb  = edge_b      + (size_t)i * 16;
        const float* sw  = src_w       + (size_t)i * HDIM * HDIM;
        const float* dw  = dst_w       + (size_t)i * HDIM * HDIM;
        const float* db  = dst_b       + (size_t)i * HDIM;
        const float* asw = attn_src_w  + (size_t)i * HDIM * HDIM;
        const float* adw = attn_dst_w  + (size_t)i * HDIM * HDIM;
        const float* aew = attn_edge_w + (size_t)i * 16 * HDIM;
        const float* gs  = gn_scale    + (size_t)i * HDIM;
        const float* go  = gn_offset   + (size_t)i * HDIM;
        const float* aw  = agg_w       + (size_t)i * HDIM * HDIM;
        const float* ab  = agg_b       + (size_t)i * HDIM;
        const float* bg  = bn_gamma    + (size_t)i * HDIM;
        const float* bb  = bn_beta     + (size_t)i * HDIM;

        // ee = relu(efeat @ edge_w + edge_b)   [E,16]
        {
            const int total = (E >> 4) * 1;
            gemm_k8_relu_kernel<<<(total + 7) / 8, blk, 0, stream>>>(
                efeat, ew, eb, ee, E, 16);
        }

        // node-side GEMMs (WMMA f32, L2-resident operands)
        gemm_k256_kernel<<<gemm256_grid, blk, 0, stream>>>(h, asw, nullptr, nullptr, nullptr, attn_s, N);
        gemm_k256_kernel<<<gemm256_grid, blk, 0, stream>>>(h, adw, nullptr, nullptr, nullptr, attn_d, N);
        gemm_k256_kernel<<<gemm256_grid, blk, 0, stream>>>(h, sw,  nullptr, nullptr, nullptr, fsrc,   N);
        gemm_k256_kernel<<<gemm256_grid, blk, 0, stream>>>(h, dw,  db,      nullptr, nullptr, dterm,  N);

        // zero msg, mdst, msrc, sdst, ssrc (contiguous 5*NH u32)
        fill_u32_kernel<<<2048, blk, 0, stream>>>((unsigned*)msg, 0u, 5 * NH);

        // dual segment softmax + message (edge logits recomputed each pass)
        seg_max_kernel<<<E, blk, 0, stream>>>(src, dst, attn_s, attn_d, ee, aew, mdst, msrc);
        seg_sum_kernel<<<E, blk, 0, stream>>>(src, dst, attn_s, attn_d, ee, aew, mdst, msrc, sdst, ssrc);
        message_kernel<<<E, blk, 0, stream>>>(src, dst, attn_s, attn_d, ee, aew, mdst, msrc, sdst, ssrc, fsrc, msg);

        // per-head GroupNorm (one wave32 == one head)
        groupnorm_kernel<<<N, blk, 0, stream>>>(msg, gs, go);

        // pre-BN sum: hlast = msg @ agg_w + agg_b + dterm (+ old hlast)
        gemm_k256_kernel<<<gemm256_grid, blk, 0, stream>>>(
            msg, aw, ab, dterm, (i > 0) ? hlast : nullptr, hlast, N);

        // BatchNorm over nodes + ReLU; final layer writes straight to d_out
        fill_u32_kernel<<<1, blk, 0, stream>>>((unsigned*)bsum, 0u, 2 * HDIM);
        bn_stats_kernel<<<256, blk, 0, stream>>>(hlast, bsum, bsq, N);
        float* bn_out = (i == L - 1) ? (float*)d_out : h;
        bn_apply_kernel<<<256, blk, 0, stream>>>(hlast, bsum, bsq, bg, bb, bn_out, N);
    }
}